// Model_88957362634934
// MI455X (gfx1250) — compile-verified
//
#include <hip/hip_runtime.h>
#include <hip/hip_bf16.h>
#include <math.h>

// ---------------------------------------------------------------------------
// LBS model for MI455X (gfx1250).
//   Outputs (flat, concatenated):
//     [0]                    verts_batched : B*V*3
//     [B*V*3]                pose          : 69
//     [+69]                  joints_out    : 69
//     [+1]                   scale         : 1
//     [+3]                   displacement  : 3
//     [...]                  local_adjust  : V*3
//
// Roofline: ~176 MB of HBM traffic (92 MB skinning reads dominate), ~0.6
// GFLOP -> memory bound (~8 us floor @ 23.3 TB/s). The (Vx23)@(23x16) blend
// goes through V_WMMA_F32_16X16X4_F32 so VALU is not the bottleneck.
// ---------------------------------------------------------------------------

#define NJ 23

typedef float v2f __attribute__((ext_vector_type(2)));
typedef float v8f __attribute__((ext_vector_type(8)));

__device__ __constant__ int c_par[NJ] =
    {-1, 0, 1, 1, 3, 4, 5, 4, 7, 4, 9, 1, 11, 12, 13, 12, 15, 12, 17, 0, 19, 0, 21};

// Hardware tanh (CDNA5 TRANS32 op). v_nop satisfies the TRANS32 hazard rule
// (one independent op or nop must come before the result is used) since the compiler
// cannot see through the asm to schedule the delay itself.
__device__ __forceinline__ float htanh(float x)
{
    float r;
    asm volatile("v_tanh_f32 %0, %1\n\tv_nop" : "=v"(r) : "v"(x));
    return r;
}

// ---------------------------------------------------------------------------
// Kernel 1: serial joint math (J=23 is tiny). One thread does everything.
//   ws[0..383]   : A^T, wsA[m*24 + j] = A_j[m] (m = flattened 4x4 row-major),
//                  column j=23 zero-padded (K padded 23 -> 24 for WMMA).
//   ws[384..386] : combined offset  loc + 0.1*tanh(displacement)
//   ws[387]      : 0.0035 * scale
// ---------------------------------------------------------------------------
__global__ void setup_kernel(const float* __restrict__ joints_rest,
                             const float* __restrict__ j0,  const float* __restrict__ j1,
                             const float* __restrict__ j2,  const float* __restrict__ j3,
                             const float* __restrict__ j4,  const float* __restrict__ j5,
                             const float* __restrict__ j12, const float* __restrict__ j13,
                             const float* __restrict__ displacement,
                             const float* __restrict__ scale,
                             const float* __restrict__ loc,
                             float* __restrict__ ws,
                             float* __restrict__ out_pose,
                             float* __restrict__ out_joints,
                             float* __restrict__ out_scale,
                             float* __restrict__ out_disp)
{
    if (threadIdx.x != 0 || blockIdx.x != 0) return;

    const float PI = 3.14159265358979323846f;
    float pose[NJ][3];
    for (int j = 0; j < NJ; ++j)
        for (int a = 0; a < 3; ++a) pose[j][a] = 0.0f;

    for (int a = 0; a < 3; ++a) {
        pose[0][a]  = (PI / 2.0f) * tanhf(j0[a]);
        pose[1][a]  = (PI / 4.0f) * tanhf(j1[a]);
        pose[2][a]  = (PI / 9.0f) * tanhf(j2[a]);
        float p3    = (PI / 3.0f) * tanhf(j3[a]);
        pose[3][a]  = p3;
        pose[11][a] = p3 * ((a == 0) ? 1.0f : -1.0f);
        pose[4][a]  = (PI / 3.0f) * tanhf(j4[a]);
        pose[12][a] = (PI / 3.0f) * tanhf(j12[a]);
        pose[13][a] = (PI / 3.0f) * tanhf(j13[a]);
        pose[5][a]  = (PI / 3.0f) * tanhf(j5[a]);
    }

    // Forward kinematics: G[j] = G[parent] @ T_local[j]
    float G[NJ][16];
    for (int j = 0; j < NJ; ++j) {
        float rx = pose[j][0], ry = pose[j][1], rz = pose[j][2];
        float ang = sqrtf(rx * rx + ry * ry + rz * rz + 1e-12f);
        float ax = rx / ang, ay = ry / ang, az = rz / ang;
        float s = sinf(ang), c = cosf(ang), mc = 1.0f - c;
        // R = I + s*K + (1-c)*K^2, K = [[0,-z,y],[z,0,-x],[-y,x,0]]
        float R[9];
        R[0] = 1.0f - mc * (ay * ay + az * az);
        R[1] = -s * az + mc * ax * ay;
        R[2] =  s * ay + mc * ax * az;
        R[3] =  s * az + mc * ax * ay;
        R[4] = 1.0f - mc * (ax * ax + az * az);
        R[5] = -s * ax + mc * ay * az;
        R[6] = -s * ay + mc * ax * az;
        R[7] =  s * ax + mc * ay * az;
        R[8] = 1.0f - mc * (ax * ax + ay * ay);

        int p = c_par[j];
        float relx = joints_rest[j * 3 + 0];
        float rely = joints_rest[j * 3 + 1];
        float relz = joints_rest[j * 3 + 2];
        if (p >= 0) {
            relx -= joints_rest[p * 3 + 0];
            rely -= joints_rest[p * 3 + 1];
            relz -= joints_rest[p * 3 + 2];
        }
        float T[16] = { R[0], R[1], R[2], relx,
                        R[3], R[4], R[5], rely,
                        R[6], R[7], R[8], relz,
                        0.0f, 0.0f, 0.0f, 1.0f };
        if (p < 0) {
            for (int k = 0; k < 16; ++k) G[j][k] = T[k];
        } else {
            for (int r = 0; r < 4; ++r)
                for (int cc = 0; cc < 4; ++cc) {
                    float acc = 0.0f;
                    for (int k = 0; k < 4; ++k) acc += G[p][r * 4 + k] * T[k * 4 + cc];
                    G[j][r * 4 + cc] = acc;
                }
        }
    }

    float disp[3], off[3];
    for (int a = 0; a < 3; ++a) disp[a] = 0.1f * tanhf(displacement[a]);
    float sf = 0.0035f * scale[0];
    for (int a = 0; a < 3; ++a) off[a] = loc[a] + disp[a];

    // A_j = G_j with translation corrected by -R_j @ joints_rest[j]; store A^T.
    for (int j = 0; j < NJ; ++j) {
        float jx = joints_rest[j * 3 + 0];
        float jy = joints_rest[j * 3 + 1];
        float jz = joints_rest[j * 3 + 2];
        float A[16];
        for (int k = 0; k < 16; ++k) A[k] = G[j][k];
        A[3]  -= G[j][0] * jx + G[j][1] * jy + G[j][2]  * jz;
        A[7]  -= G[j][4] * jx + G[j][5] * jy + G[j][6]  * jz;
        A[11] -= G[j][8] * jx + G[j][9] * jy + G[j][10] * jz;
        for (int m = 0; m < 16; ++m) ws[m * 24 + j] = A[m];

        for (int a = 0; a < 3; ++a) {
            out_pose[j * 3 + a]   = pose[j][a];
            out_joints[j * 3 + a] = sf * G[j][a * 4 + 3] + off[a];
        }
    }
    for (int m = 0; m < 16; ++m) ws[m * 24 + 23] = 0.0f;

    ws[384] = off[0]; ws[385] = off[1]; ws[386] = off[2]; ws[387] = sf;

    out_scale[0] = scale[0];
    out_disp[0] = displacement[0];
    out_disp[1] = displacement[1];
    out_disp[2] = displacement[2];
}

// ---------------------------------------------------------------------------
// Kernel 2: per-vertex skinning. One wave32 handles 16-vertex tiles in a
// wave-uniform grid-stride loop (EXEC all-ones at every WMMA):
//   D(16x16) = A^T(16x24) x S^T(24x16)  via 6x V_WMMA_F32_16X16X4_F32.
//   Rows of D = 16 Tv elements, cols = vertices. Lane n (n<16) holds vertex n
//   elements 0..7 (Tv rows 0,1); lane n+16 holds elements 8..15 (rows 2,3).
//   Operand layout (32-bit, K=4 chunk): lanes 0-15 -> K={0,1} in vgpr{0,1},
//   lanes 16-31 -> K={2,3}, mirroring the ISA's 16x4 f32 A-matrix table.
// vt/la tile loads and all vertex stores go through LDS so global traffic is
// coalesced b128; cross-lane LDS ordering enforced with s_wait_dscnt 0.
// NOTE: assumes nverts % 16 == 0 (holds: V = 1,000,000).
// ---------------------------------------------------------------------------
__global__ __launch_bounds__(256) void lbs_kernel(
    const float* __restrict__ skin,       // V x 23
    const float* __restrict__ vt,         // V x 3
    const float* __restrict__ la,         // V x 3
    const float* __restrict__ ws,         // A^T + scalars
    float* __restrict__ out_verts,        // B x V x 3
    float* __restrict__ out_la,           // V x 3 passthrough
    int nverts, int nbatch)
{
    __shared__ float sVT[8][48];
    __shared__ float sLA[8][48];
    __shared__ float sOUT[8][48];

    const int lane = threadIdx.x & 31;
    const int wv   = threadIdx.x >> 5;    // wave within block (blockDim = 256)
    const int half = lane >> 4;
    const int m    = lane & 15;           // A-operand row (Tv element index)
    const int vl   = lane & 15;           // vertex slot within tile

    // Hoisted per-wave constants: A^T operand fragments + scale/offset.
    v2f aop[6];
#pragma unroll
    for (int c = 0; c < 6; ++c) {
        const int k0 = 4 * c + 2 * half;
        aop[c].x = ws[m * 24 + k0];
        aop[c].y = ws[m * 24 + k0 + 1];
    }
    const float off0 = ws[384], off1 = ws[385], off2 = ws[386], sf = ws[387];

    const int ntiles = nverts >> 4;       // exact: nverts % 16 == 0
    const int nwaves = (int)(gridDim.x * blockDim.x) >> 5;
    const int wave0  = (blockIdx.x * (int)blockDim.x + (int)threadIdx.x) >> 5;

    for (int tile = wave0; tile < ntiles; tile += nwaves) {
        const int vbase = tile << 4;
        const int vIdx  = vbase + vl;

        // Prefetch next grid-stride tile's skinning rows into cache.
        {
            int nt = tile + nwaves;
            if (nt >= ntiles) nt = tile;
            __builtin_prefetch(skin + (size_t)(nt << 4) * NJ, 0, 1);
        }

        // ---- Coalesced b128 stage-in of vt/la tile (+ la passthrough) ----
        if (lane < 12) {
            const float4 t = ((const float4*)(vt + (size_t)vbase * 3))[lane];
            const float4 u = ((const float4*)(la + (size_t)vbase * 3))[lane];
            ((float4*)sVT[wv])[lane] = t;
            ((float4*)sLA[wv])[lane] = u;
            ((float4*)(out_la + (size_t)vbase * 3))[lane] = u;   // passthrough
        }
        asm volatile("s_wait_dscnt 0x0" ::: "memory");
        __builtin_amdgcn_wave_barrier();

        // ---- Blend: 6 chained fp32 WMMAs over K = 24 (23 + zero pad) ----
        v8f acc = {0.f, 0.f, 0.f, 0.f, 0.f, 0.f, 0.f, 0.f};
#pragma unroll
        for (int c = 0; c < 6; ++c) {
            const int k0 = 4 * c + 2 * half;      // k0 <= 22 always
            int k1 = k0 + 1;
            const int k1c = (k1 < NJ) ? k1 : (NJ - 1);   // clamp, then select 0
            v2f b;
            b.x = skin[(size_t)vIdx * NJ + k0];
            const float t = skin[(size_t)vIdx * NJ + k1c];
            b.y = (k1 < NJ) ? t : 0.0f;
            acc = __builtin_amdgcn_wmma_f32_16x16x4_f32(
                /*neg_a=*/false, aop[c], /*neg_b=*/false, b,
                /*c_mod=*/(short)0, acc, /*reuse_a=*/false, /*reuse_b=*/false);
        }

        // ---- Per-vertex epilogue (hardware v_tanh_f32, branch-free) ----
        const float px = sVT[wv][vl * 3 + 0] + 0.1f * htanh(sLA[wv][vl * 3 + 0]);
        const float py = sVT[wv][vl * 3 + 1] + 0.1f * htanh(sLA[wv][vl * 3 + 1]);
        const float pz = sVT[wv][vl * 3 + 2] + 0.1f * htanh(sLA[wv][vl * 3 + 2]);

        const float r0 = acc[0] * px + acc[1] * py + acc[2] * pz + acc[3];
        const float r1 = acc[4] * px + acc[5] * py + acc[6] * pz + acc[7];

        if (half == 0) {
            sOUT[wv][vl * 3 + 0] = sf * r0 + off0;   // Tv row 0 -> x
            sOUT[wv][vl * 3 + 1] = sf * r1 + off1;   // Tv row 1 -> y
        } else {
            sOUT[wv][vl * 3 + 2] = sf * r0 + off2;   // Tv row 2 -> z
        }
        asm volatile("s_wait_dscnt 0x0" ::: "memory");
        __builtin_amdgcn_wave_barrier();

        // ---- Coalesced b128 batched stores ----
        if (lane < 12) {
            const float4 o = ((const float4*)sOUT[wv])[lane];
            for (int b2 = 0; b2 < nbatch; ++b2) {
                ((float4*)(out_verts + (size_t)b2 * nverts * 3 +
                           (size_t)vbase * 3))[lane] = o;
            }
        }
    }
}

// ---------------------------------------------------------------------------
extern "C" void kernel_launch(void* const* d_in, const int* in_sizes, int n_in,
                              void* d_out, int out_size, void* d_ws, size_t ws_size,
                              hipStream_t stream)
{
    const float* v_template   = (const float*)d_in[0];
    const float* skinning     = (const float*)d_in[1];
    const float* joints_rest  = (const float*)d_in[2];
    const float* j0           = (const float*)d_in[3];
    const float* j1           = (const float*)d_in[4];
    const float* j2           = (const float*)d_in[5];
    const float* j3           = (const float*)d_in[6];
    const float* j4           = (const float*)d_in[7];
    const float* j5           = (const float*)d_in[8];
    const float* j12          = (const float*)d_in[9];
    const float* j13          = (const float*)d_in[10];
    const float* local_adjust = (const float*)d_in[11];
    const float* displacement = (const float*)d_in[12];
    const float* scale        = (const float*)d_in[13];
    const float* estloc       = (const float*)d_in[14];

    const int V = in_sizes[0] / 3;
    // out_size = B*3V + 69 + 69 + 1 + 3 + 3V
    const int B = (out_size - 142 - 3 * V) / (3 * V);

    float* out        = (float*)d_out;
    float* out_verts  = out;
    float* out_pose   = out + (size_t)B * 3 * V;
    float* out_joints = out_pose + 3 * NJ;
    float* out_scale  = out_joints + 3 * NJ;
    float* out_disp   = out_scale + 1;
    float* out_la     = out_disp + 3;
    float* ws         = (float*)d_ws;

    setup_kernel<<<1, 32, 0, stream>>>(joints_rest, j0, j1, j2, j3, j4, j5, j12, j13,
                                       displacement, scale, estloc, ws,
                                       out_pose, out_joints, out_scale, out_disp);

    const int tiles = V >> 4;                         // 16 vertices per tile
    // ~4 tiles per wave, 8 waves per block.
    int waves  = (tiles + 3) / 4;
    int blocks = (waves + 7) / 8;
    if (blocks < 1) blocks = 1;
    lbs_kernel<<<blocks, 256, 0, stream>>>(skinning, v_template, local_adjust, ws,
                                           out_verts, out_la, V, B);
}